// radon_forward_42984032698900
// MI455X (gfx1250) — compile-verified
//
#include <hip/hip_runtime.h>
#include <math.h>

// ---------------------------------------------------------------------------
// Radon forward projection via type-2 NUFFT on MI455X (gfx1250):
//  * fft2 of the 1024x1024 padded grid  -> DFT matmuls on f32 WMMA
//  * final length-768 ifft + shifts + Re -> DFT matmul on f32 WMMA
//  * KB 6x6 gridding gather              -> scalar VALU kernel (L2-resident)
// GEMM: block = 8 waves = 128(M)x16(N) C-strip; B K-panel staged in LDS
// (K-major, so the WMMA B pair is one ds_load_b64), A fed by contiguous
// float2 global loads with global_prefetch_b8 ahead of the pipeline.
// ---------------------------------------------------------------------------

typedef __attribute__((ext_vector_type(2))) float v2f;
typedef __attribute__((ext_vector_type(8))) float v8f;

#define JT 6
#define ALPHA_KB 14.04f            /* 2.34 * J */
#define HH 512
#define WW 512
#define GG 1024
#define NANG 720
#define NDET 768
#define KSAMP (NANG * NDET)
#define MPAD 768                   /* detector-space GEMM padded to 768 rows */
#define PI_F 3.14159265358979323846f

#define KC 32                      /* K-panel staged in LDS per iteration */
#define LDB (KC + 2)               /* K-major LDS stride, pad kills conflicts */

// Abramowitz & Stegun 9.8.1 / 9.8.2 polynomial I0 (matches np.i0 to ~1e-7 rel)
__device__ __forceinline__ float bessel_i0f(float x) {
  float ax = fabsf(x);
  if (ax < 3.75f) {
    float t = ax * (1.0f / 3.75f);
    t *= t;
    return 1.0f + t * (3.5156229f + t * (3.0899424f + t * (1.2067492f +
                 t * (0.2659732f + t * (0.0360768f + t * 0.0045813f)))));
  } else {
    float t = 3.75f / ax;
    float p = 0.39894228f + t * (0.01328592f + t * (0.00225319f +
              t * (-0.00157565f + t * (0.00916281f + t * (-0.02057706f +
              t * (0.02635537f + t * (-0.01647633f + t * 0.00392377f)))))));
    return expf(ax) * rsqrtf(ax) * p;
  }
}

// 1 / FT(KB kernel) apodization factor for pixel n of an N-long axis, grid G
__device__ __forceinline__ float apod_inv(int n, int N, int G, float i0a) {
  float u = (float)(n - N / 2) / (float)G;
  float pj = PI_F * (float)JT * u;
  float arg = ALPHA_KB * ALPHA_KB - pj * pj;
  float s = sqrtf(fabsf(arg) + 1e-20f);
  float F = (arg > 0.0f) ? (0.5f * (expf(s) - expf(-s)) / s) : (sinf(s) / s);
  F *= ((float)JT / i0a);
  return 1.0f / F;
}

// ---- twiddle matrix for 1024-pt forward DFT: W[u,m] = exp(-2*pi*i*u*m/G) ---
__global__ void k_fill_W(float* __restrict__ Wr, float* __restrict__ Wi) {
  int idx = blockIdx.x * blockDim.x + threadIdx.x;
  if (idx >= GG * GG) return;
  int u = idx >> 10, m = idx & (GG - 1);
  int p = (u * m) & (GG - 1);               // exact phase mod G (G power of 2)
  float ph = -2.0f * PI_F * (float)p * (1.0f / (float)GG);
  Wr[idx] = cosf(ph);
  Wi[idx] = sinf(ph);
}

// ---- B[m,n] for the fused (ifftshift -> ifft -> Re -> fftshift) stage ------
// out[a,n] = sum_m kdr[a,m]*Br[m,n] - kdi[a,m]*Bi[m,n]
// B[m,n] = (-1)^{n'} * exp(+2*pi*i*m*n'/ND)/ND,  n' = (n+ND/2) mod ND
__global__ void k_fill_B(float* __restrict__ Br, float* __restrict__ Bi) {
  int idx = blockIdx.x * blockDim.x + threadIdx.x;
  if (idx >= NDET * NDET) return;
  int m = idx / NDET, n = idx - m * NDET;
  int np = (n + NDET / 2) % NDET;
  int p = (m * np) % NDET;                  // exact integer phase reduction
  float ph = 2.0f * PI_F * (float)p * (1.0f / (float)NDET);
  float sc = ((np & 1) ? -1.0f : 1.0f) * (1.0f / (float)NDET);
  Br[idx] = sc * cosf(ph);
  Bi[idx] = sc * sinf(ph);
}

// ---- apodize + zero-pad + ifftshift folded: Xr = shifted padded image -----
__global__ void k_fill_grid(const float* __restrict__ img, float* __restrict__ Xr) {
  int idx = blockIdx.x * blockDim.x + threadIdx.x;
  if (idx >= GG * GG) return;
  int i = idx >> 10, j = idx & (GG - 1);
  int r = (i < 256) ? (i + 256) : ((i >= 768) ? (i - 768) : -1);
  int c = (j < 256) ? (j + 256) : ((j >= 768) ? (j - 768) : -1);
  float v = 0.0f;
  if (r >= 0 && c >= 0) {
    float i0a = bessel_i0f(ALPHA_KB);
    v = img[r * WW + c] * apod_inv(r, HH, GG, i0a) * apod_inv(c, WW, GG, i0a);
  }
  Xr[idx] = v;
}

// ---- f32 WMMA GEMM: C = alpha*A(MxK)*B(KxN) + beta*C ----------------------
// Grid: (N/16, M/128). Block: 256 threads = 8 waves, each wave one 16x16
// tile stacked in M; all share one 16-wide B strip staged through LDS.
// M, N multiples of 128/16 and K multiple of KC -> every wave fully active
// (WMMA requires EXEC all-ones).
__global__ void __launch_bounds__(256)
k_gemm_wmma(const float* __restrict__ A, const float* __restrict__ B,
            float* __restrict__ C, int M, int N, int K,
            float alpha, float beta) {
  __shared__ float bs[16 * LDB];            // B panel, K-major per column

  int wv = threadIdx.x >> 5;                // wave in block: M sub-tile
  int lane = threadIdx.x & 31;
  int hh = lane >> 4;                       // half-wave: K pair {0,1} / {2,3}
  int lm = lane & 15;

  int tN = blockIdx.x << 4;                 // 16-col strip
  int tM = (blockIdx.y << 7) + (wv << 4);   // this wave's 16 rows

  const float* Ap = A + (size_t)(tM + lm) * K + (hh ? 2 : 0);
  const float* Bp = B + (size_t)(threadIdx.x >> 4) * N + tN + (threadIdx.x & 15);

  v8f acc = {};
  for (int kc = 0; kc < K; kc += KC) {
    // cooperative stage: 256 threads x 2 elems = KC x 16 panel, K-major in LDS
    {
      int e0 = threadIdx.x;                 // row = e>>4 in [0,16), col = e&15
      int e1 = threadIdx.x + 256;
      bs[(e0 & 15) * LDB + (e0 >> 4)] = Bp[(size_t)kc * N];
      bs[(e1 & 15) * LDB + (e1 >> 4)] = Bp[(size_t)(kc + 16) * N];
    }
    __syncthreads();
    __builtin_prefetch(Ap + kc + KC, 0, 0); // global_prefetch_b8 next A panel
#pragma unroll
    for (int kk = 0; kk < KC; kk += 4) {
      v2f a = *(const v2f*)(Ap + kc + kk);  // A[m][k0..k0+1], ISA 16x4 layout
      v2f b = *(const v2f*)&bs[lm * LDB + kk + (hh ? 2 : 0)]; // ds_load_b64
      acc = __builtin_amdgcn_wmma_f32_16x16x4_f32(
          /*neg_a=*/false, a, /*neg_b=*/false, b,
          /*c_mod=*/(short)0, acc, /*reuse_a=*/false, /*reuse_b=*/false);
    }
    __syncthreads();
  }

  // D layout: VGPR r -> rows {tM+r, tM+r+8} by half-wave, col = tN+lm
#pragma unroll
  for (int r = 0; r < 8; r++) {
    int row = tM + r + (hh ? 8 : 0);
    size_t o = (size_t)row * N + tN + lm;
    float v = alpha * acc[r];
    if (beta != 0.0f) v += beta * C[o];
    C[o] = v;
  }
}

// ---- KB 6x6 gridding gather into MPAD x NDET (rows >= NANG zeroed) --------
__global__ void k_interp(const float* __restrict__ ksp,
                         const float* __restrict__ Yr, const float* __restrict__ Yi,
                         float* __restrict__ kdr, float* __restrict__ kdi) {
  int k = blockIdx.x * blockDim.x + threadIdx.x;
  if (k >= MPAD * NDET) return;
  if (k >= KSAMP) { kdr[k] = 0.0f; kdi[k] = 0.0f; return; }  // pad rows
  float i0a = bessel_i0f(ALPHA_KB);
  float inv_i0a = 1.0f / i0a;

  float g1 = fmodf(ksp[k]         * ((float)GG / (2.0f * PI_F)), (float)GG);
  float g2 = fmodf(ksp[KSAMP + k] * ((float)GG / (2.0f * PI_F)), (float)GG);
  if (g1 < 0.0f) g1 += (float)GG;
  if (g2 < 0.0f) g2 += (float)GG;
  int b1 = (int)floorf(g1);
  int b2 = (int)floorf(g2);

  float wx[JT], wy[JT];
  int ix[JT], iy[JT];
#pragma unroll
  for (int j = 0; j < JT; j++) {
    int p1 = b1 - 2 + j;
    int p2 = b2 - 2 + j;
    float u1 = g1 - (float)p1;
    float u2 = g2 - (float)p2;
    float x1 = fmaxf(1.0f - (u1 * (1.0f / 3.0f)) * (u1 * (1.0f / 3.0f)), 0.0f);
    float x2 = fmaxf(1.0f - (u2 * (1.0f / 3.0f)) * (u2 * (1.0f / 3.0f)), 0.0f);
    wx[j] = bessel_i0f(ALPHA_KB * sqrtf(x1)) * inv_i0a;
    wy[j] = bessel_i0f(ALPHA_KB * sqrtf(x2)) * inv_i0a;
    ix[j] = (p1 + GG) & (GG - 1);
    iy[j] = (p2 + GG) & (GG - 1);
  }

  float sr = 0.0f, si = 0.0f;
#pragma unroll
  for (int a = 0; a < JT; a++) {
    int ro = ix[a] << 10;
    float wa = wx[a];
#pragma unroll
    for (int b = 0; b < JT; b++) {
      float w = wa * wy[b];
      int o = ro + iy[b];
      sr += w * Yr[o];
      si += w * Yi[o];
    }
  }
  const float sc = 1.0f / 1024.0f;          // fft2 ortho norm 1/sqrt(G1*G2)
  kdr[k] = sr * sc;
  kdi[k] = si * sc;
}

// ---- crop padded 768x768 result to the 720x768 output ---------------------
__global__ void k_crop(const float* __restrict__ Cpad, float* __restrict__ out) {
  int k = blockIdx.x * blockDim.x + threadIdx.x;
  if (k < KSAMP) out[k] = Cpad[k];          // row-major: first 720 rows
}

// ---------------------------------------------------------------------------
extern "C" void kernel_launch(void* const* d_in, const int* in_sizes, int n_in,
                              void* d_out, int out_size, void* d_ws, size_t ws_size,
                              hipStream_t stream) {
  const float* img = (const float*)d_in[0];   // (1,1,512,512) f32
  const float* ksp = (const float*)d_in[1];   // (2, K) f32
  float* out = (float*)d_out;                 // (720,768) f32
  float* ws = (float*)d_ws;

  const size_t NG = (size_t)GG * GG;          // 1,048,576
  const size_t NB = (size_t)NDET * NDET;      //   589,824

  float* Wr   = ws;            float* Wi   = Wr   + NG;
  float* Xr   = Wi   + NG;     float* Tr   = Xr   + NG;
  float* Ti   = Tr   + NG;     float* Yr   = Ti   + NG;
  float* Yi   = Yr   + NG;
  float* Br   = Yi   + NG;     float* Bi   = Br   + NB;
  float* kdr  = Bi   + NB;     float* kdi  = kdr  + NB;  // padded 768x768
  float* Cpad = kdi  + NB;                                // ~41 MB total

  auto gemm = [&](const float* A, const float* B, float* C,
                  int M, int N, int K, float alpha, float beta) {
    k_gemm_wmma<<<dim3(N / 16, M / 128), dim3(256), 0, stream>>>(A, B, C,
                                                                 M, N, K,
                                                                 alpha, beta);
  };

  // Twiddles + apodized shifted grid
  k_fill_W   <<<dim3((GG * GG) / 256),     dim3(256), 0, stream>>>(Wr, Wi);
  k_fill_B   <<<dim3((NDET * NDET) / 256), dim3(256), 0, stream>>>(Br, Bi);
  k_fill_grid<<<dim3((GG * GG) / 256),     dim3(256), 0, stream>>>(img, Xr);

  // fft2(X) = W * X * W  (X real -> 6 real GEMMs)
  gemm(Wr, Xr, Tr, GG, GG, GG,  1.0f, 0.0f);  // Tr = Wr*X
  gemm(Wi, Xr, Ti, GG, GG, GG,  1.0f, 0.0f);  // Ti = Wi*X
  gemm(Tr, Wr, Yr, GG, GG, GG,  1.0f, 0.0f);  // Yr = Tr*Wr
  gemm(Ti, Wi, Yr, GG, GG, GG, -1.0f, 1.0f);  // Yr -= Ti*Wi
  gemm(Tr, Wi, Yi, GG, GG, GG,  1.0f, 0.0f);  // Yi = Tr*Wi
  gemm(Ti, Wr, Yi, GG, GG, GG,  1.0f, 1.0f);  // Yi += Ti*Wr

  // KB gridding gather (rows 720..767 zero-padded)
  k_interp<<<dim3((MPAD * NDET) / 256), dim3(256), 0, stream>>>(ksp, Yr, Yi,
                                                                kdr, kdi);

  // fused ifftshift/ifft/Re/fftshift along detectors (padded to 768 rows)
  gemm(kdr, Br, Cpad, MPAD, NDET, NDET,  1.0f, 0.0f);
  gemm(kdi, Bi, Cpad, MPAD, NDET, NDET, -1.0f, 1.0f);

  // crop to (720, 768)
  k_crop<<<dim3(KSAMP / 256), dim3(256), 0, stream>>>(Cpad, out);
}